// LowRankRNN_76141180223841
// MI455X (gfx1250) — compile-verified
//
#include <hip/hip_runtime.h>
#include <hip/hip_bf16.h>

typedef __attribute__((ext_vector_type(2))) float v2f;
typedef __attribute__((ext_vector_type(8))) float v8f;

#define HID   8192
#define BATCHN 512
#define RANKN 8
#define NIN   16
#define NOUT  4

// Native CDNA5 v_tanh_f32 (TRANS32 op) when available; avoids the branchy
// OCML tanh expansion (EXEC divergence + ~20 VALU ops) in the hot loop.
__device__ __forceinline__ float dev_tanh(float x) {
#if __has_builtin(__builtin_amdgcn_tanhf)
  return __builtin_amdgcn_tanhf(x);
#else
  return tanhf(x);
#endif
}

// ---------------------------------------------------------------------------
// Kernel 0: zero the output accumulator region of d_out (512 x 4 floats).
// ---------------------------------------------------------------------------
__global__ void __launch_bounds__(256) rnn_zero_kernel(float* __restrict__ out) {
  int i = blockIdx.x * blockDim.x + threadIdx.x;
  if (i < BATCHN * NOUT) out[i] = 0.0f;
}

// ---------------------------------------------------------------------------
// Kernel 1: u[b][r] = (1/H) * sum_j tanh(h[b][j]) * R[r][j]
// One block (256 threads) per batch row; LDS tree reduction. R (256 KB) stays
// resident in the 192 MB L2 across all 512 blocks.
// ---------------------------------------------------------------------------
__global__ void __launch_bounds__(256) rnn_u_kernel(const float* __restrict__ h,
                                                    const float* __restrict__ R,
                                                    float* __restrict__ u) {
  const int b   = blockIdx.x;
  const int tid = threadIdx.x;
  const float* hrow = h + (size_t)b * HID;

  float acc[RANKN];
#pragma unroll
  for (int r = 0; r < RANKN; ++r) acc[r] = 0.0f;

  for (int j = tid; j < HID; j += 256) {
    float th = dev_tanh(hrow[j]);
#pragma unroll
    for (int r = 0; r < RANKN; ++r)
      acc[r] += th * R[(size_t)r * HID + j];
  }

  __shared__ float red[256];
#pragma unroll
  for (int r = 0; r < RANKN; ++r) {
    red[tid] = acc[r];
    __syncthreads();
    for (int s = 128; s > 0; s >>= 1) {
      if (tid < s) red[tid] += red[tid + s];
      __syncthreads();
    }
    if (tid == 0) u[b * RANKN + r] = red[0] * (1.0f / (float)HID);
    __syncthreads();
  }
}

// ---------------------------------------------------------------------------
// Kernel 2: fused Euler step + readout partials.
// Each wave owns a 16(batch) x 128(H) strip = 8 tiles of 16x16.
// Per tile: c = input_tile @ W_in_tile^T  (4x V_WMMA_F32_16X16X4_F32, K=16)
//             + u_tile     @ L_tile^T     (2x V_WMMA_F32_16X16X4_F32, K=8)
// Then per C-fragment element: new_h = h + dt*(c - h); accumulate
// tanh(new_h)*W_out partials in registers; shfl-reduce + atomicAdd at the end.
// ---------------------------------------------------------------------------
__global__ void __launch_bounds__(256) rnn_step_kernel(
    const float* __restrict__ input, const float* __restrict__ h,
    const float* __restrict__ W_in,  const float* __restrict__ L,
    const float* __restrict__ u,     const float* __restrict__ W_out,
    float* __restrict__ new_h,       float* __restrict__ out) {
  const int tid  = threadIdx.x;
  const int wave = tid >> 5;
  const int lane = tid & 31;
  const int half = lane >> 4;   // K/M split: lanes 0-15 vs 16-31
  const int ln   = lane & 15;

  const int b0 = blockIdx.y * 16;                 // 16 batch rows per tile
  const int jw = blockIdx.x * 1024 + wave * 128;  // this wave's 128-col strip

  // ---- A operands, loaded once for the whole strip ----
  // A-matrix 16x4 f32 layout: lane(half,ln) holds {A[ln][2*half], A[ln][2*half+1]}
  const int br = b0 + ln;
  const v2f* inrow = (const v2f*)(input + (size_t)br * NIN);
  v2f a_in[4];
#pragma unroll
  for (int i = 0; i < 4; ++i) a_in[i] = inrow[2 * i + half];   // K = 4i+2*half
  const v2f* urow = (const v2f*)(u + (size_t)br * RANKN);
  v2f a_u[2];
#pragma unroll
  for (int i = 0; i < 2; ++i) a_u[i] = urow[2 * i + half];

  float acc_out[8][NOUT];
#pragma unroll
  for (int p = 0; p < 8; ++p)
#pragma unroll
    for (int o = 0; o < NOUT; ++o) acc_out[p][o] = 0.0f;

  for (int t = 0; t < 8; ++t) {
    const int jt = jw + t * 16;
    const int j  = jt + ln;   // this lane's column in the C/D fragment

    // B-matrix 4x16 f32 layout: lane(half,ln) holds {B[2*half][ln], B[2*half+1][ln]}
    // B[k][n] = W_in[jt+n][k]  (resp. L[jt+n][k]) -> contiguous float2 loads.
    const v2f* wrow = (const v2f*)(W_in + (size_t)(jt + ln) * NIN);
    const v2f* lrow = (const v2f*)(L    + (size_t)(jt + ln) * RANKN);

    v8f c = {0.f, 0.f, 0.f, 0.f, 0.f, 0.f, 0.f, 0.f};
    // ff tile: K = 16 -> 4 WMMAs
#pragma unroll
    for (int i = 0; i < 4; ++i)
      c = __builtin_amdgcn_wmma_f32_16x16x4_f32(
          false, a_in[i], false, wrow[2 * i + half], (short)0, c, false, false);
    // recurrent tile: K = rank = 8 -> 2 WMMAs (u already carries the 1/H scale)
#pragma unroll
    for (int i = 0; i < 2; ++i)
      c = __builtin_amdgcn_wmma_f32_16x16x4_f32(
          false, a_u[i], false, lrow[2 * i + half], (short)0, c, false, false);

    // Elementwise Euler step on the C fragment.
    // C/D layout: VGPR p, lane(half,ln) -> element [row = b0+p+8*half][col = j]
#pragma unroll
    for (int p = 0; p < 8; ++p) {
      const int row    = b0 + p + 8 * half;
      const size_t idx = (size_t)row * HID + j;
      if (t + 1 < 8) __builtin_prefetch(&h[idx + 16], 0, 3);  // next tile's h
      float hv = h[idx];
      float nh = hv + 0.1f * (c[p] - hv);   // h + dt*(-h + ff + v)/tau, tau=1
      new_h[idx] = nh;
      float tn = dev_tanh(nh);
#pragma unroll
      for (int o = 0; o < NOUT; ++o)
        acc_out[p][o] += tn * W_out[(size_t)o * HID + j];
    }
  }

  // Reduce readout partials over the 16 lanes of each half-wave, then one
  // scaled atomic per (row, out).
#pragma unroll
  for (int p = 0; p < 8; ++p) {
#pragma unroll
    for (int o = 0; o < NOUT; ++o) {
      float v = acc_out[p][o];
      v += __shfl_xor(v, 1);
      v += __shfl_xor(v, 2);
      v += __shfl_xor(v, 4);
      v += __shfl_xor(v, 8);
      if (ln == 0) {
        const int row = b0 + p + 8 * half;
        atomicAdd(&out[row * NOUT + o], v * (1.0f / (float)HID));
      }
    }
  }
}

// ---------------------------------------------------------------------------
extern "C" void kernel_launch(void* const* d_in, const int* in_sizes, int n_in,
                              void* d_out, int out_size, void* d_ws, size_t ws_size,
                              hipStream_t stream) {
  (void)in_sizes; (void)n_in; (void)out_size; (void)ws_size;
  const float* input = (const float*)d_in[0];   // [512, 16]
  const float* h     = (const float*)d_in[1];   // [512, 8192]
  const float* W_in  = (const float*)d_in[2];   // [8192, 16]
  const float* L     = (const float*)d_in[3];   // [8192, 8]
  const float* R     = (const float*)d_in[4];   // [8, 8192]
  const float* W_out = (const float*)d_in[5];   // [4, 8192]

  float* out   = (float*)d_out;                     // [512,4] output first
  float* new_h = out + (size_t)BATCHN * NOUT;       // then [512,8192] new_hidden
  float* u     = (float*)d_ws;                      // [512,8] scratch

  rnn_zero_kernel<<<(BATCHN * NOUT + 255) / 256, 256, 0, stream>>>(out);
  rnn_u_kernel<<<BATCHN, 256, 0, stream>>>(h, R, u);
  rnn_step_kernel<<<dim3(HID / 1024, BATCHN / 16), 256, 0, stream>>>(
      input, h, W_in, L, u, W_out, new_h, out);
}